// InfoNCELoss_32856499814917
// MI455X (gfx1250) — compile-verified
//
#include <hip/hip_runtime.h>
#include <math.h>

// ---------------------------------------------------------------------------
// InfoNCE loss, fused: never materializes the 8192x8192 sim matrix.
//   k1: normalize rows (fold 1/sqrt(T) so WMMA output == sim), zero accum
//   k2: fp32 WMMA (v_wmma_f32_16x16x4_f32) 16x16 sim tiles + fused exp/pos
//   k3: per-row log + reduction
//   k4: final scalar
// ---------------------------------------------------------------------------

typedef float v2f __attribute__((ext_vector_type(2)));
typedef float v8f __attribute__((ext_vector_type(8)));

#define B_ROWS 8192
#define DIM 256
#define TILE 16
#define STRIDE (DIM + 4)                 // 260 floats: row stride -> 4-bank skew, no LDS conflicts
#define NTILES (B_ROWS / TILE)           // 512 column tiles
#define CSPLIT 8                         // column split for parallelism
#define TILES_PER_BLK (NTILES / CSPLIT)  // 64
#define WAVES 2                          // 2 waves/block -> ~50KB static LDS (safe)

__global__ __launch_bounds__(256) void nce_normalize(const float* __restrict__ emb,
                                                     float* __restrict__ En,
                                                     float* __restrict__ gExp,
                                                     float* __restrict__ gPos,
                                                     float* __restrict__ gCnt,
                                                     float* __restrict__ accum) {
  __shared__ float red[256];
  const int row = blockIdx.x;
  const int t = threadIdx.x;
  const float v = emb[(size_t)row * DIM + t];
  red[t] = v * v;
  __syncthreads();
#pragma unroll
  for (int s = 128; s > 0; s >>= 1) {
    if (t < s) red[t] += red[t + s];
    __syncthreads();
  }
  const float nrm = sqrtf(red[0]);
  // 1/max(||x||,1e-12) * 1/sqrt(T), T=0.1 -> sqrt(10)
  const float scale = 3.16227766016838f / fmaxf(nrm, 1e-12f);
  En[(size_t)row * DIM + t] = v * scale;
  if (t == 0) { gExp[row] = 0.f; gPos[row] = 0.f; gCnt[row] = 0.f; }
  if (row == 0 && t == 0) { accum[0] = 0.f; accum[1] = 0.f; }
}

__global__ __launch_bounds__(WAVES * 32) void nce_sim(const float* __restrict__ En,
                                                      const int* __restrict__ ids,
                                                      float* __restrict__ gExp,
                                                      float* __restrict__ gPos,
                                                      float* __restrict__ gCnt) {
  __shared__ float ldsA[TILE * STRIDE];          // 16 strip rows
  __shared__ float ldsB[WAVES][TILE * STRIDE];   // per-wave column tile
  __shared__ float sE[TILE], sP[TILE], sC[TILE];

  const int tid = threadIdx.x;
  const int lane = tid & 31;
  const int wave = tid >> 5;
  const int ibase = blockIdx.x * TILE;           // 16-row strip
  const int tbase = blockIdx.y * TILES_PER_BLK;  // this block's column-tile range

  if (tid < TILE) { sE[tid] = 0.f; sP[tid] = 0.f; sC[tid] = 0.f; }

  // cooperative load of the A strip: 16 rows x 256 floats, float4 coalesced
  for (int idx = tid; idx < TILE * (DIM / 4); idx += WAVES * 32) {
    const int r = idx >> 6, c4 = idx & 63;
    const float4 v = *(const float4*)(En + (size_t)(ibase + r) * DIM + c4 * 4);
    *(float4*)&ldsA[r * STRIDE + c4 * 4] = v;
  }
  __syncthreads();

  // wave32 WMMA f32 16x16x4 operand layout:
  //   A: lane l holds row (l&15), K = kbase + 2*(l>>4) .. +1
  //   B: lane n holds col (n&15) (== row of En), same K pattern
  //   C/D: VGPR r, lane l -> M = r + 8*(l>>4), N = l&15
  const int hi = lane >> 4;
  const int lo = lane & 15;
  const int rowOff = lo * STRIDE + 2 * hi;

  int idr[8];
#pragma unroll
  for (int r = 0; r < 8; ++r) idr[r] = ids[ibase + r + 8 * hi];

  float accE[8], accP[8], accC[8];
#pragma unroll
  for (int r = 0; r < 8; ++r) { accE[r] = 0.f; accP[r] = 0.f; accC[r] = 0.f; }

  float* myB = ldsB[wave];

  for (int jt = tbase + wave; jt < tbase + TILES_PER_BLK; jt += WAVES) {
    const int jbase = jt * TILE;
    // stage this wave's B tile (16 rows x 256 floats); same-wave DS ordering
    // makes store->load safe without a barrier.
    for (int idx = lane; idx < TILE * (DIM / 4); idx += 32) {
      const int r = idx >> 6, c4 = idx & 63;
      const float4 v = *(const float4*)(En + (size_t)(jbase + r) * DIM + c4 * 4);
      *(float4*)&myB[r * STRIDE + c4 * 4] = v;
    }

    v8f c = {0.f, 0.f, 0.f, 0.f, 0.f, 0.f, 0.f, 0.f};
#pragma unroll 16
    for (int k = 0; k < DIM; k += 4) {
      const v2f a = *(const v2f*)&ldsA[rowOff + k];
      const v2f b = *(const v2f*)&myB[rowOff + k];
      // (neg_a, A, neg_b, B, c_mod, C, reuse_a, reuse_b)
      c = __builtin_amdgcn_wmma_f32_16x16x4_f32(false, a, false, b, (short)0, c,
                                                false, false);
    }

    const int gj = jbase + lo;
    const int idc = ids[gj];
#pragma unroll
    for (int r = 0; r < 8; ++r) {
      const int gi = ibase + r + 8 * hi;
      const float s = c[r];
      if (gi != gj) {
        accE[r] += __expf(s);  // |s| <= 10: no max-shift needed in fp32
        if (idr[r] == idc) { accP[r] += s; accC[r] += 1.f; }
      }
    }
  }

  // fold lane partials into per-row LDS accumulators
#pragma unroll
  for (int r = 0; r < 8; ++r) {
    const int m = r + 8 * hi;
    atomicAdd(&sE[m], accE[r]);
    atomicAdd(&sP[m], accP[r]);
    atomicAdd(&sC[m], accC[r]);
  }
  __syncthreads();
  if (tid < TILE) {
    atomicAdd(&gExp[ibase + tid], sE[tid]);
    atomicAdd(&gPos[ibase + tid], sP[tid]);
    atomicAdd(&gCnt[ibase + tid], sC[tid]);
  }
}

__global__ __launch_bounds__(256) void nce_rowreduce(const float* __restrict__ gExp,
                                                     const float* __restrict__ gPos,
                                                     const float* __restrict__ gCnt,
                                                     float* __restrict__ accum) {
  __shared__ float rv[256], rc[256];
  const int t = threadIdx.x;
  const int i = blockIdx.x * 256 + t;
  const float cnt = gCnt[i];
  const float denom = __logf(gExp[i]);  // logsumexp over j != i
  rv[t] = cnt * denom - gPos[i];        // sum over positives of (denom_i - sim_ij)
  rc[t] = cnt;
  __syncthreads();
#pragma unroll
  for (int s = 128; s > 0; s >>= 1) {
    if (t < s) { rv[t] += rv[t + s]; rc[t] += rc[t + s]; }
    __syncthreads();
  }
  if (t == 0) {
    atomicAdd(&accum[0], rv[0]);
    atomicAdd(&accum[1], rc[0]);
  }
}

__global__ void nce_finalize(const float* __restrict__ accum, float* __restrict__ out) {
  const float c = accum[1];
  out[0] = (c > 0.f) ? (accum[0] / fmaxf(c, 1.f)) : 0.f;
}

extern "C" void kernel_launch(void* const* d_in, const int* in_sizes, int n_in,
                              void* d_out, int out_size, void* d_ws, size_t ws_size,
                              hipStream_t stream) {
  (void)in_sizes; (void)n_in; (void)out_size; (void)ws_size;
  const float* emb = (const float*)d_in[0];
  const int* ids = (const int*)d_in[1];

  float* En = (float*)d_ws;                       // 8192*256 fp32 = 8 MB
  float* gExp = En + (size_t)B_ROWS * DIM;        // 8192 fp32
  float* gPos = gExp + B_ROWS;                    // 8192 fp32
  float* gCnt = gPos + B_ROWS;                    // 8192 fp32
  float* accum = gCnt + B_ROWS;                   // 2 fp32

  nce_normalize<<<B_ROWS, 256, 0, stream>>>(emb, En, gExp, gPos, gCnt, accum);

  dim3 grid(NTILES, CSPLIT);
  nce_sim<<<grid, WAVES * 32, 0, stream>>>(En, ids, gExp, gPos, gCnt);

  nce_rowreduce<<<B_ROWS / 256, 256, 0, stream>>>(gExp, gPos, gCnt, accum);
  nce_finalize<<<1, 1, 0, stream>>>(accum, (float*)d_out);
}